// MixtureOfDepths_40355512714062
// MI455X (gfx1250) — compile-verified
//
#include <hip/hip_runtime.h>
#include <hip/hip_bf16.h>

typedef __attribute__((ext_vector_type(2))) float v2f;
typedef __attribute__((ext_vector_type(4))) float v4f;
typedef __attribute__((ext_vector_type(8))) float v8f;

#define B_    4
#define T_    8192
#define D_    1024
#define F_    4096
#define CAP_  4096
#define FCHUNK 128

__device__ __forceinline__ float fast_tanh(float z) {
#if __has_builtin(__builtin_amdgcn_tanhf)
    return __builtin_amdgcn_tanhf(z);       // v_tanh_f32 (gfx1250 TRANS op)
#else
    // branch-free: tanh(z) = 1 - 2/(1+exp(2z)); __expf -> v_exp_f32
    float e = __expf(2.0f * z);
    return 1.0f - 2.0f / (e + 1.0f);
#endif
}

__device__ __forceinline__ float gelu_tanh(float x) {
    const float c0 = 0.7978845608028654f;   // sqrt(2/pi)
    const float c1 = 0.044715f;
    float x3 = x * x * x;
    return 0.5f * x * (1.0f + fast_tanh(c0 * (x + c1 * x3)));
}

// ---------------- Kernel 1: router scores (one wave32 per token) ----------------
__global__ void __launch_bounds__(256)
mod_scores(const float* __restrict__ x, const float* __restrict__ wr,
           float* __restrict__ scores) {
    int wave = (blockIdx.x * blockDim.x + threadIdx.x) >> 5;
    int lane = threadIdx.x & 31;
    if (wave >= B_ * T_) return;
    const float* row = x + (size_t)wave * D_;
    float s = 0.f;
    #pragma unroll 4
    for (int i = lane; i < D_; i += 32) s += row[i] * wr[i];
    #pragma unroll
    for (int off = 16; off > 0; off >>= 1) s += __shfl_xor(s, off, 32);
    if (lane == 0) scores[wave] = s;
}

// ------- Kernel 2: exact top-capacity selection per batch row (1 block/row) -------
__global__ void __launch_bounds__(1024)
mod_select(const float* __restrict__ scores, int* __restrict__ idx_list) {
    const int b = blockIdx.x;
    const int t = threadIdx.x;          // 0..1023
    const int NPT = T_ / 1024;          // 8 scores per thread
    __shared__ int cnt;
    __shared__ int appendCnt;
    __shared__ int scan[1024];

    unsigned key[NPT];
    int tok[NPT];
    #pragma unroll
    for (int i = 0; i < NPT; ++i) {
        int tk = t + i * 1024;
        tok[i] = tk;
        unsigned u = __float_as_uint(scores[b * T_ + tk]);
        // order-preserving map: larger float -> larger uint
        key[i] = (u & 0x80000000u) ? ~u : (u | 0x80000000u);
    }

    // binary search: thr = max v such that count(key >= v) >= CAP_
    unsigned lo = 0u, hi = 0xFFFFFFFFu;
    while (lo < hi) {
        unsigned mid = (unsigned)(((unsigned long long)lo + (unsigned long long)hi + 1ull) >> 1);
        if (t == 0) cnt = 0;
        __syncthreads();
        int c = 0;
        #pragma unroll
        for (int i = 0; i < NPT; ++i) c += (key[i] >= mid) ? 1 : 0;
        atomicAdd(&cnt, c);
        __syncthreads();
        int total = cnt;
        __syncthreads();
        if (total >= CAP_) lo = mid; else hi = mid - 1;
    }
    const unsigned thr = lo;

    if (t == 0) { cnt = 0; appendCnt = 0; }
    __syncthreads();
    int cgt_local = 0, ceq_local = 0;
    #pragma unroll
    for (int i = 0; i < NPT; ++i) {
        cgt_local += (key[i] > thr) ? 1 : 0;
        ceq_local += (key[i] == thr) ? 1 : 0;
    }
    atomicAdd(&cnt, cgt_local);
    scan[t] = ceq_local;
    __syncthreads();
    const int c_gt = cnt;
    // Hillis-Steele inclusive scan over per-thread equal counts (deterministic ranks)
    for (int off = 1; off < 1024; off <<= 1) {
        int v = (t >= off) ? scan[t - off] : 0;
        __syncthreads();
        scan[t] += v;
        __syncthreads();
    }
    int eq_rank = scan[t] - ceq_local;      // exclusive base for this thread
    const int need = CAP_ - c_gt;
    int* list = idx_list + b * CAP_;
    #pragma unroll
    for (int i = 0; i < NPT; ++i) {
        if (key[i] > thr) {
            int slot = atomicAdd(&appendCnt, 1);    // order irrelevant, set deterministic
            list[slot] = tok[i];
        } else if (key[i] == thr) {
            if (eq_rank < need) list[c_gt + eq_rank] = tok[i];
            eq_rank++;
        }
    }
}

// --------- Kernel 3: fused FFN with f32 WMMA, tile = 16 tokens per block ---------
__global__ void __launch_bounds__(256)
mod_ffn(const float* __restrict__ x, const float* __restrict__ w1,
        const float* __restrict__ b1, const float* __restrict__ w2,
        const float* __restrict__ b2, const int* __restrict__ idx_list,
        float* __restrict__ out) {
    __shared__ float lds_x[16 * D_];        // 64 KB: gathered X tile (row-major)
    __shared__ float lds_h[16 * FCHUNK];    //  8 KB: gelu(H) chunk
    __shared__ int   lds_tok[16];

    const int b      = blockIdx.y;
    const int tile   = blockIdx.x;          // 0..CAP_/16-1
    const int tid    = threadIdx.x;
    const int wave   = tid >> 5;            // 0..7
    const int lane   = tid & 31;
    const int lane16 = lane & 15;
    const int hi     = lane >> 4;           // 0 | 1
    const int koff   = hi * 2;              // K sub-offset for f32 WMMA operands

    if (tid < 16) lds_tok[tid] = idx_list[b * CAP_ + tile * 16 + tid];
    __syncthreads();

    // gather X tile (16 x 1024) into LDS, float4 granularity, coalesced
    #pragma unroll 4
    for (int row = 0; row < 16; ++row) {
        int col = tid * 4;
        const float* src = x + ((size_t)b * T_ + (size_t)lds_tok[row]) * D_ + col;
        *(v4f*)&lds_x[row * D_ + col] = *(const v4f*)src;
    }
    __syncthreads();

    const int out_col_base = wave * 128;    // this wave owns 128 output columns
    v8f zero = {};
    v8f acc[8];
    #pragma unroll
    for (int j = 0; j < 8; ++j) acc[j] = zero;

    for (int ch = 0; ch < F_ / FCHUNK; ++ch) {
        const int fbase = ch * FCHUNK;
        const int hcol  = fbase + wave * 16 + lane16;   // F column of this wave's H tile

        // ---- GEMM1: H[16, 16] = X[16,1024] @ W1[:, hcol block], K = 1024 ----
        v8f hacc = zero;
        #pragma unroll 4
        for (int k = 0; k < D_; k += 4) {
            v2f a = *(const v2f*)&lds_x[lane16 * D_ + k + koff];
            v2f bv;
            bv.x = w1[(size_t)(k + koff)     * F_ + hcol];
            bv.y = w1[(size_t)(k + koff + 1) * F_ + hcol];
            hacc = __builtin_amdgcn_wmma_f32_16x16x4_f32(
                false, a, false, bv, (short)0, hacc, false, false);
        }

        // bias + gelu (C/D layout: VGPR r holds row r + 8*hi, col = lane16)
        float b1v = b1[hcol];
        float hval[8];
        #pragma unroll
        for (int r = 0; r < 8; ++r) hval[r] = gelu_tanh(hacc[r] + b1v);

        __syncthreads();   // previous chunk's GEMM2 done reading lds_h
        #pragma unroll
        for (int r = 0; r < 8; ++r)
            lds_h[(r + hi * 8) * FCHUNK + wave * 16 + lane16] = hval[r];
        __syncthreads();   // lds_h visible

        // ---- GEMM2: O[16, 128] += gelu(H)[16,128] @ W2[chunk, wave's 128 cols] ----
        #pragma unroll 2
        for (int k = 0; k < FCHUNK; k += 4) {
            v2f a = *(const v2f*)&lds_h[lane16 * FCHUNK + k + koff];
            const float* w2p = w2 + (size_t)(fbase + k + koff) * D_ + out_col_base + lane16;
            #pragma unroll
            for (int j = 0; j < 8; ++j) {
                v2f bv;
                bv.x = w2p[j * 16];
                bv.y = w2p[D_ + j * 16];
                acc[j] = __builtin_amdgcn_wmma_f32_16x16x4_f32(
                    false, a, false, bv, (short)0, acc[j], false, false);
            }
        }
    }

    // epilogue: + b2, scatter rows by token index
    #pragma unroll
    for (int j = 0; j < 8; ++j) {
        int col = out_col_base + j * 16 + lane16;
        float b2v = b2[col];
        #pragma unroll
        for (int r = 0; r < 8; ++r) {
            int row = r + hi * 8;
            out[((size_t)b * T_ + (size_t)lds_tok[row]) * D_ + col] = acc[j][r] + b2v;
        }
    }
}

extern "C" void kernel_launch(void* const* d_in, const int* in_sizes, int n_in,
                              void* d_out, int out_size, void* d_ws, size_t ws_size,
                              hipStream_t stream) {
    const float* x  = (const float*)d_in[0];
    const float* wr = (const float*)d_in[1];
    const float* w1 = (const float*)d_in[2];
    const float* b1 = (const float*)d_in[3];
    const float* w2 = (const float*)d_in[4];
    const float* b2 = (const float*)d_in[5];
    float* out = (float*)d_out;

    float* scores   = (float*)d_ws;                                     // B*T floats
    int*   idx_list = (int*)((char*)d_ws + (size_t)B_ * T_ * sizeof(float)); // B*CAP ints

    // passthrough: out = x (selected rows overwritten by mod_ffn)
    hipMemcpyAsync(out, x, (size_t)B_ * T_ * D_ * sizeof(float),
                   hipMemcpyDeviceToDevice, stream);

    mod_scores<<<dim3((B_ * T_) / 8), dim3(256), 0, stream>>>(x, wr, scores);
    mod_select<<<dim3(B_), dim3(1024), 0, stream>>>(scores, idx_list);
    mod_ffn<<<dim3(CAP_ / 16, B_), dim3(256), 0, stream>>>(x, w1, b1, w2, b2, idx_list, out);
}